// DualGraphLearner_4191888081049
// MI455X (gfx1250) — compile-verified
//
#include <hip/hip_runtime.h>
#include <hip/hip_bf16.h>

#define N_NODES 50000
#define N_EDGES 800000
#define IN_DIM  128
#define HEADS   2
#define DHEAD   32
#define HD      64   // HEADS * DHEAD

typedef __attribute__((ext_vector_type(2))) float v2f;
typedef __attribute__((ext_vector_type(8))) float v8f;

// ---------------- float atomic max/min via integer punning ----------------
__device__ __forceinline__ float atomicMaxFloat(float* addr, float value) {
    if (value >= 0.0f)
        return __int_as_float(atomicMax((int*)addr, __float_as_int(value)));
    else
        return __uint_as_float(atomicMin((unsigned int*)addr, __float_as_uint(value)));
}
__device__ __forceinline__ float atomicMinFloat(float* addr, float value) {
    if (value >= 0.0f)
        return __int_as_float(atomicMin((int*)addr, __float_as_int(value)));
    else
        return __uint_as_float(atomicMax((unsigned int*)addr, __float_as_uint(value)));
}

// ---------------- init scratch ----------------
__global__ __launch_bounds__(256)
void init_kernel(float* __restrict__ mmax, float* __restrict__ denom,
                 float* __restrict__ chans, float* __restrict__ minmax) {
    int i = blockIdx.x * 256 + threadIdx.x;
    if (i < N_NODES * HEADS) {
        mmax[i]  = -__builtin_inff();
        denom[i] = 0.0f;
    }
    if (i < 2 * HD) chans[i] = 0.0f;
    if (i == 0) { minmax[0] = __builtin_inff(); minmax[1] = -__builtin_inff(); }
}

// ---------------- fused projections via fp32 WMMA ----------------
// Computes q,k,v = x@W+b and accum = x@Ws+bs (skip init for the aggregator).
// grid = (N/16, 2), block = 256 (8 waves). Each wave owns one 16x16 output
// tile: nTile = blockIdx.y*8 + wave, column range [nTile*16, nTile*16+16)
// of the 256-wide concatenated [Wq|Wk|Wv|Ws] output.
__global__ __launch_bounds__(256)
void proj_wmma_kernel(const float* __restrict__ x,
                      const float* __restrict__ Wq, const float* __restrict__ bq,
                      const float* __restrict__ Wk, const float* __restrict__ bk,
                      const float* __restrict__ Wv, const float* __restrict__ bv,
                      const float* __restrict__ Ws, const float* __restrict__ bs,
                      float* __restrict__ qo, float* __restrict__ ko,
                      float* __restrict__ vo, float* __restrict__ accum) {
    __shared__ float xs[16 * IN_DIM];             // 8 KB strip of x
    const int m0  = blockIdx.x * 16;
    const int tid = threadIdx.x;

    // coalesced load of the 16x128 x strip
    for (int i = tid; i < 16 * IN_DIM; i += 256) {
        int r = i >> 7, c = i & (IN_DIM - 1);
        xs[i] = x[(size_t)(m0 + r) * IN_DIM + c];
    }
    __syncthreads();

    const int wave  = tid >> 5;
    const int lane  = tid & 31;
    const int nTile = blockIdx.y * 8 + wave;      // 0..15
    const int wsel  = nTile >> 2;                 // which weight matrix
    const int colInW = (nTile & 3) * 16;          // base column within [0,64)

    const float* W; const float* bias; float* outp;
    switch (wsel) {
        case 0:  W = Wq; bias = bq; outp = qo;    break;
        case 1:  W = Wk; bias = bk; outp = ko;    break;
        case 2:  W = Wv; bias = bv; outp = vo;    break;
        default: W = Ws; bias = bs; outp = accum; break;
    }

    // A (16x4 f32): lanes 0-15 M=0..15 K={k0,k0+1}; lanes 16-31 K={k0+2,k0+3}
    // B (4x16 f32): same K split, N = lane&15
    const int mn   = lane & 15;
    const int khi  = (lane >> 4) << 1;            // 0 or 2
    const int wcol = colInW + mn;                 // output column within weight

    v8f acc = {0.f, 0.f, 0.f, 0.f, 0.f, 0.f, 0.f, 0.f};
    #pragma unroll
    for (int k0 = 0; k0 < IN_DIM; k0 += 4) {
        const int kb = k0 + khi;
        v2f a; a.x = xs[mn * IN_DIM + kb];     a.y = xs[mn * IN_DIM + kb + 1];
        v2f b; b.x = W[(size_t)kb * HD + wcol]; b.y = W[(size_t)(kb + 1) * HD + wcol];
        acc = __builtin_amdgcn_wmma_f32_16x16x4_f32(
            /*neg_a=*/false, a, /*neg_b=*/false, b,
            /*c_mod=*/(short)0, acc, /*reuse_a=*/false, /*reuse_b=*/false);
    }

    // D layout: VGPR r holds row (r + (lane<16?0:8)), col = lane&15
    const float bval  = bias[wcol];
    const int rowOff  = (lane < 16) ? 0 : 8;
    #pragma unroll
    for (int r = 0; r < 8; ++r) {
        int m = m0 + r + rowOff;
        outp[(size_t)m * HD + wcol] = acc[r] + bval;
    }
}

// ---------------- per-edge attention logits + segment max ----------------
__global__ __launch_bounds__(256)
void edge_alpha_kernel(const int* __restrict__ ei,
                       const float* __restrict__ q, const float* __restrict__ k,
                       float* __restrict__ alpha, float* __restrict__ mmax) {
    int idx = blockIdx.x * 256 + threadIdx.x;
    if (idx >= N_EDGES * HEADS) return;
    int e = idx >> 1, h = idx & 1;
    int src = ei[e], dst = ei[N_EDGES + e];
    const float4* qp = (const float4*)(q + (size_t)dst * HD + h * DHEAD);
    const float4* kp = (const float4*)(k + (size_t)src * HD + h * DHEAD);
    float s = 0.f;
    #pragma unroll
    for (int i = 0; i < DHEAD / 4; ++i) {
        float4 a = qp[i], b = kp[i];
        s += a.x * b.x + a.y * b.y + a.z * b.z + a.w * b.w;
    }
    s *= 0.17677669529663687f;  // 1/sqrt(32)
    alpha[idx] = s;
    atomicMaxFloat(&mmax[dst * HEADS + h], s);
}

// ---------------- exp(alpha - max) + segment sum ----------------
__global__ __launch_bounds__(256)
void edge_exp_kernel(const int* __restrict__ ei,
                     float* __restrict__ alpha,
                     const float* __restrict__ mmax,
                     float* __restrict__ denom) {
    int idx = blockIdx.x * 256 + threadIdx.x;
    if (idx >= N_EDGES * HEADS) return;
    int e = idx >> 1, h = idx & 1;
    int dst = ei[N_EDGES + e];
    float a = __expf(alpha[idx] - mmax[dst * HEADS + h]);
    alpha[idx] = a;
    atomicAdd(&denom[dst * HEADS + h], a);
}

// ---------------- weighted scatter-add of values ----------------
// 64 threads per edge (one per output channel); accum was pre-seeded with skip.
__global__ __launch_bounds__(256)
void edge_aggr_kernel(const int* __restrict__ ei,
                      const float* __restrict__ v,
                      const float* __restrict__ alpha,
                      const float* __restrict__ denom,
                      float* __restrict__ accum) {
    int t = blockIdx.x * 256 + threadIdx.x;
    int e = t >> 6;
    if (e >= N_EDGES) return;
    int c = t & 63, h = c >> 5;
    int src = ei[e], dst = ei[N_EDGES + e];
    float a = alpha[e * HEADS + h];
    float w = a / (denom[dst * HEADS + h] + 1e-16f);
    atomicAdd(&accum[(size_t)dst * HD + c], w * v[(size_t)src * HD + c]);
}

// ---------------- per-channel sum / sumsq reduction ----------------
__global__ __launch_bounds__(256)
void stats_kernel(const float* __restrict__ accum, float* __restrict__ chans) {
    __shared__ float ls[2 * HD];
    int tid = threadIdx.x;
    if (tid < 2 * HD) ls[tid] = 0.f;
    __syncthreads();
    float s = 0.f, s2 = 0.f;
    int c = (blockIdx.x * 256 + tid) & 63;  // stride is multiple of 64 -> fixed c
    for (int i = blockIdx.x * 256 + tid; i < N_NODES * HD; i += gridDim.x * 256) {
        float xv = accum[i];
        s += xv; s2 += xv * xv;
    }
    atomicAdd(&ls[c], s);
    atomicAdd(&ls[HD + c], s2);
    __syncthreads();
    if (tid < 2 * HD) atomicAdd(&chans[tid], ls[tid]);
}

// ---------------- GraphNorm scale/shift per channel ----------------
__global__ void finalize_stats_kernel(const float* __restrict__ chans,
                                      const float* __restrict__ gn_w,
                                      const float* __restrict__ gn_b,
                                      const float* __restrict__ gn_ms,
                                      float* __restrict__ ss) {
    int c = threadIdx.x;
    if (c >= HD) return;
    const float inv_n = 1.0f / (float)N_NODES;
    float mean = chans[c] * inv_n;
    float ex2  = chans[HD + c] * inv_n;
    float al   = gn_ms[c];
    // var of (out - al*mean) = E[out^2] - 2*al*mean*E[out] + al^2*mean^2
    float var  = ex2 - 2.f * al * mean * mean + al * al * mean * mean;
    float rstd = rsqrtf(var + 1e-5f);
    ss[c]      = rstd * gn_w[c];
    ss[HD + c] = gn_b[c] - al * mean * rstd * gn_w[c];
}

// ---------------- norm + relu, write out, global min/max ----------------
__global__ __launch_bounds__(256)
void norm_relu_kernel(const float* __restrict__ accum,
                      const float* __restrict__ ss,
                      float* __restrict__ out, float* __restrict__ minmax) {
    float lmin = __builtin_inff(), lmax = -__builtin_inff();
    int tid = threadIdx.x;
    for (int i = blockIdx.x * 256 + tid; i < N_NODES * HD; i += gridDim.x * 256) {
        int c = i & 63;
        float y  = accum[i] * ss[c] + ss[HD + c];
        float xt = fmaxf(y, 0.f);
        out[i] = xt;
        lmin = fminf(lmin, xt);
        lmax = fmaxf(lmax, xt);
    }
    #pragma unroll
    for (int off = 16; off > 0; off >>= 1) {
        lmin = fminf(lmin, __shfl_xor(lmin, off, 32));
        lmax = fmaxf(lmax, __shfl_xor(lmax, off, 32));
    }
    if ((tid & 31) == 0) {
        atomicMinFloat(&minmax[0], lmin);
        atomicMaxFloat(&minmax[1], lmax);
    }
}

// ---------------- final global min-max rescale ----------------
__global__ __launch_bounds__(256)
void final_scale_kernel(float* __restrict__ out, const float* __restrict__ minmax) {
    int i = blockIdx.x * 256 + threadIdx.x;
    if (i >= N_NODES * HD) return;
    float mn = minmax[0], mx = minmax[1];
    out[i] = (out[i] - mn) / (mx - mn + 1e-8f);
}

// ---------------- launch ----------------
extern "C" void kernel_launch(void* const* d_in, const int* in_sizes, int n_in,
                              void* d_out, int out_size, void* d_ws, size_t ws_size,
                              hipStream_t stream) {
    const float* x    = (const float*)d_in[0];
    const int*   ei   = (const int*)d_in[1];
    const float* Wq   = (const float*)d_in[2];
    const float* bq   = (const float*)d_in[3];
    const float* Wk   = (const float*)d_in[4];
    const float* bk   = (const float*)d_in[5];
    const float* Wv   = (const float*)d_in[6];
    const float* bv   = (const float*)d_in[7];
    const float* Ws   = (const float*)d_in[8];
    const float* bs   = (const float*)d_in[9];
    const float* gn_w = (const float*)d_in[10];
    const float* gn_b = (const float*)d_in[11];
    const float* gn_ms= (const float*)d_in[12];
    float* out = (float*)d_out;

    // workspace carving (all float32)
    float* ws = (float*)d_ws;
    size_t off = 0;
    float* q      = ws + off; off += (size_t)N_NODES * HD;   // 3.2M
    float* k      = ws + off; off += (size_t)N_NODES * HD;
    float* v      = ws + off; off += (size_t)N_NODES * HD;
    float* accum  = ws + off; off += (size_t)N_NODES * HD;   // seeded with skip
    float* alpha  = ws + off; off += (size_t)N_EDGES * HEADS;
    float* mmax   = ws + off; off += (size_t)N_NODES * HEADS;
    float* denom  = ws + off; off += (size_t)N_NODES * HEADS;
    float* chans  = ws + off; off += 2 * HD;                 // sum | sumsq
    float* ss     = ws + off; off += 2 * HD;                 // scale | shift
    float* minmax = ws + off; off += 2;

    // 1. init scratch that is accumulated into
    init_kernel<<<(N_NODES * HEADS + 255) / 256, 256, 0, stream>>>(mmax, denom, chans, minmax);

    // 2. fused fp32-WMMA projections (q,k,v + skip into accum)
    dim3 pgrid(N_NODES / 16, 2);
    proj_wmma_kernel<<<pgrid, 256, 0, stream>>>(x, Wq, bq, Wk, bk, Wv, bv, Ws, bs,
                                                q, k, v, accum);

    // 3. edge attention logits + segment max
    int eh_blocks = (N_EDGES * HEADS + 255) / 256;
    edge_alpha_kernel<<<eh_blocks, 256, 0, stream>>>(ei, q, k, alpha, mmax);

    // 4. exp + segment sum
    edge_exp_kernel<<<eh_blocks, 256, 0, stream>>>(ei, alpha, mmax, denom);

    // 5. normalized weighted scatter-add of v on top of skip
    int aggr_blocks = (N_EDGES * HD + 255) / 256;
    edge_aggr_kernel<<<aggr_blocks, 256, 0, stream>>>(ei, v, alpha, denom, accum);

    // 6. GraphNorm statistics
    stats_kernel<<<1024, 256, 0, stream>>>(accum, chans);
    finalize_stats_kernel<<<1, 64, 0, stream>>>(chans, gn_w, gn_b, gn_ms, ss);

    // 7. norm + relu + global min/max
    norm_relu_kernel<<<1024, 256, 0, stream>>>(accum, ss, out, minmax);

    // 8. final min-max rescale
    final_scale_kernel<<<(N_NODES * HD + 255) / 256, 256, 0, stream>>>(out, minmax);
}